// HPNLayer_86148454023234
// MI455X (gfx1250) — compile-verified
//
#include <hip/hip_runtime.h>
#include <math.h>

// HPN layer for MI455X (gfx1250, wave32).
// GEMMs use V_WMMA_F32_16X16X4_F32 (full fp32 precision — GEMM is not the
// bottleneck; the edge scatter/gather at ~5GB of L2-resident traffic is).

typedef __attribute__((ext_vector_type(2))) float v2f;
typedef __attribute__((ext_vector_type(8))) float v8f;

#define DDIM 128
#define ALPHA_F 0.1f

// ---------------------------------------------------------------- utilities
__global__ __launch_bounds__(256) void zero_kernel(float* __restrict__ p, long n) {
  long i = (long)blockIdx.x * blockDim.x + threadIdx.x;
  if (i < n) p[i] = 0.0f;
}

__global__ __launch_bounds__(256) void zero4_kernel(float4* __restrict__ p, long n4) {
  long i = (long)blockIdx.x * blockDim.x + threadIdx.x;
  if (i < n4) p[i] = make_float4(0.f, 0.f, 0.f, 0.f);
}

// deg_out over src, deg_in over dst (float atomics; counts are exact in fp32)
__global__ __launch_bounds__(256) void degree_kernel(const int* __restrict__ src,
                                                     const int* __restrict__ dst,
                                                     float* __restrict__ deg_s,
                                                     float* __restrict__ deg_d, int nE) {
  int e = blockIdx.x * blockDim.x + threadIdx.x;
  if (e < nE) {
    atomicAdd(deg_s + src[e], 1.0f);
    atomicAdd(deg_d + dst[e], 1.0f);
  }
}

// v = clip(deg,1,inf)^-0.5, in place
__global__ __launch_bounds__(256) void norm_kernel(float* __restrict__ v, long n) {
  long i = (long)blockIdx.x * blockDim.x + threadIdx.x;
  if (i < n) v[i] = rsqrtf(fmaxf(v[i], 1.0f));
}

// ------------------------------------------------- WMMA GEMM: relu(A@W + b)
// One wave -> one 16x16 tile of C. Block = 256 thr = 8 waves = one 16-row
// strip across all 8 column tiles (D=128).
__global__ __launch_bounds__(256) void gemm_relu_kernel(
    const float* __restrict__ A, const float* __restrict__ W,
    const float* __restrict__ bias, float* __restrict__ Cout, int nrows) {
  const int lane = threadIdx.x & 31;
  const int wave = threadIdx.x >> 5;
  const int m0 = blockIdx.x * 16;
  const int n0 = wave * 16;
  const int t  = lane & 15;          // A row / B,C column within the tile
  const int kh = (lane >> 4) << 1;   // 0 or 2: K sub-offset for this lane half

  const int arow_i = min(m0 + t, nrows - 1);      // keep EXEC uniform
  const float* arow = A + (long)arow_i * DDIM;
  const float* wcol = W + n0 + t;

  v8f acc = {};
  for (int k = 0; k < DDIM; k += 4) {
    v2f a = *(const v2f*)(arow + k + kh);         // A[m][k+kh], A[m][k+kh+1]
    v2f b;
    b.x = wcol[(long)(k + kh) * DDIM];            // W[k+kh  ][col]
    b.y = wcol[(long)(k + kh + 1) * DDIM];        // W[k+kh+1][col]
    acc = __builtin_amdgcn_wmma_f32_16x16x4_f32(
        false, a, false, b, (short)0, acc, false, false);
  }

  const int col   = n0 + t;
  const int rbase = m0 + ((lane >> 4) << 3);      // rows: rbase..rbase+7
  const float bv  = bias[col];
#pragma unroll
  for (int r = 0; r < 8; ++r) {
    int row = rbase + r;
    if (row < nrows) {
      float v = acc[r] + bv;
      Cout[(long)row * DDIM + col] = v > 0.f ? v : 0.f;
    }
  }
}

// -------------------- WMMA GEMM fused with tanh() . q, summed over all rows
__global__ __launch_bounds__(256) void attn_kernel(
    const float* __restrict__ Z, const float* __restrict__ W,
    const float* __restrict__ bias, const float* __restrict__ q,
    float* __restrict__ wsum, int nrows) {
  const int lane = threadIdx.x & 31;
  const int wave = threadIdx.x >> 5;
  const int m0 = blockIdx.x * 16;
  const int n0 = wave * 16;
  const int t  = lane & 15;
  const int kh = (lane >> 4) << 1;

  const int arow_i = min(m0 + t, nrows - 1);
  const float* arow = Z + (long)arow_i * DDIM;
  const float* wcol = W + n0 + t;

  v8f acc = {};
  for (int k = 0; k < DDIM; k += 4) {
    v2f a = *(const v2f*)(arow + k + kh);
    v2f b;
    b.x = wcol[(long)(k + kh) * DDIM];
    b.y = wcol[(long)(k + kh + 1) * DDIM];
    acc = __builtin_amdgcn_wmma_f32_16x16x4_f32(
        false, a, false, b, (short)0, acc, false, false);
  }

  const int col   = n0 + t;
  const int rbase = m0 + ((lane >> 4) << 3);
  const float bv  = bias[col];
  const float qv  = q[col];
  float part = 0.0f;
#pragma unroll
  for (int r = 0; r < 8; ++r)
    if (rbase + r < nrows) part += tanhf(acc[r] + bv) * qv;

#pragma unroll
  for (int off = 16; off > 0; off >>= 1) part += __shfl_xor(part, off, 32);
  if (lane == 0) atomicAdd(wsum, part);
}

// ---------------------------------------------- APPNP edge scatter (1 wave/edge)
// agg[dst] += x[src] * norm_s[src]; each lane moves 4 contiguous floats.
__global__ __launch_bounds__(256) void scatter_kernel(
    const float* __restrict__ x, const float* __restrict__ norm_s,
    const int* __restrict__ src, const int* __restrict__ dst,
    float* __restrict__ agg, int nE) {
  long gid = (long)blockIdx.x * blockDim.x + threadIdx.x;
  int e = (int)(gid >> 5);
  if (e >= nE) return;
  int lane = (int)(gid & 31);
  int s = src[e];
  int d = dst[e];
  float ns = norm_s[s];
  float4 v = *(const float4*)(x + (long)s * DDIM + lane * 4);
  float* o = agg + (long)d * DDIM + lane * 4;
  atomicAdd(o + 0, v.x * ns);
  atomicAdd(o + 1, v.y * ns);
  atomicAdd(o + 2, v.z * ns);
  atomicAdd(o + 3, v.w * ns);
}

// x = (1-a)*agg*norm_d[node] + a*h0   (float4 over N*32 vectors)
__global__ __launch_bounds__(256) void combine_kernel(
    const float4* __restrict__ agg, const float* __restrict__ norm_d,
    const float4* __restrict__ h0, float4* __restrict__ xo, int nrows) {
  long i = (long)blockIdx.x * blockDim.x + threadIdx.x;
  if (i >= (long)nrows * (DDIM / 4)) return;
  int node = (int)(i >> 5);  // D/4 == 32 float4 per row
  float nd = norm_d[node] * (1.0f - ALPHA_F);
  float4 a = agg[i];
  float4 h = h0[i];
  xo[i] = make_float4(a.x * nd + ALPHA_F * h.x, a.y * nd + ALPHA_F * h.y,
                      a.z * nd + ALPHA_F * h.z, a.w * nd + ALPHA_F * h.w);
}

// beta = softmax(wsum * 1/N) over the 2 metapaths
__global__ void softmax2_kernel(const float* __restrict__ wsum,
                                float* __restrict__ beta, float invN) {
  float w0 = wsum[0] * invN, w1 = wsum[1] * invN;
  float m = fmaxf(w0, w1);
  float e0 = __expf(w0 - m), e1 = __expf(w1 - m);
  float inv = 1.0f / (e0 + e1);
  beta[0] = e0 * inv;
  beta[1] = e1 * inv;
}

__global__ __launch_bounds__(256) void mix_kernel(
    const float4* __restrict__ z0, const float4* __restrict__ z1,
    const float* __restrict__ beta, float4* __restrict__ out, long n4) {
  long i = (long)blockIdx.x * blockDim.x + threadIdx.x;
  if (i >= n4) return;
  float b0 = beta[0], b1 = beta[1];
  float4 a = z0[i], b = z1[i];
  out[i] = make_float4(b0 * a.x + b1 * b.x, b0 * a.y + b1 * b.y,
                       b0 * a.z + b1 * b.z, b0 * a.w + b1 * b.w);
}

// ----------------------------------------------------------------- launcher
static inline int cdivl(long a, long b) { return (int)((a + b - 1) / b); }

extern "C" void kernel_launch(void* const* d_in, const int* in_sizes, int n_in,
                              void* d_out, int out_size, void* d_ws, size_t ws_size,
                              hipStream_t stream) {
  const float* h     = (const float*)d_in[0];
  const int*   src0  = (const int*)d_in[1];
  const int*   dst0  = (const int*)d_in[2];
  const int*   src1  = (const int*)d_in[3];
  const int*   dst1  = (const int*)d_in[4];
  const float* W_h   = (const float*)d_in[5];
  const float* b_h   = (const float*)d_in[6];
  const float* W_att = (const float*)d_in[7];
  const float* b_att = (const float*)d_in[8];
  const float* q_att = (const float*)d_in[9];
  float* out = (float*)d_out;

  const int  N  = in_sizes[0] / DDIM;
  const int  E  = in_sizes[1];
  const long ND = (long)N * DDIM;

  // Workspace layout (floats): h1 | z0 | z1 | agg | ns0 nd0 ns1 nd1 | wsum[2] | beta[2]
  float* ws   = (float*)d_ws;
  float* h1   = ws;
  float* z0   = h1 + ND;
  float* z1   = z0 + ND;
  float* agg  = z1 + ND;
  float* ns0  = agg + ND;
  float* nd0  = ns0 + N;
  float* ns1  = nd0 + N;
  float* nd1  = ns1 + N;
  float* wsum = nd1 + N;   // 2 floats
  float* beta = wsum + 2;  // 2 floats

  dim3 blk(256);

  // degrees -> symmetric norms (zero ns0..nd1 and wsum in one pass)
  zero_kernel<<<cdivl(4L * N + 2, 256), blk, 0, stream>>>(ns0, 4L * N + 2);
  degree_kernel<<<cdivl(E, 256), blk, 0, stream>>>(src0, dst0, ns0, nd0, E);
  degree_kernel<<<cdivl(E, 256), blk, 0, stream>>>(src1, dst1, ns1, nd1, E);
  norm_kernel<<<cdivl(4L * N, 256), blk, 0, stream>>>(ns0, 4L * N);

  // h1 = relu(h @ W_h + b_h)   [WMMA fp32]
  gemm_relu_kernel<<<cdivl(N, 16), blk, 0, stream>>>(h, W_h, b_h, h1, N);

  // APPNP, 3 steps per metapath; x ping-pongs through z_p (h0 == h1)
  const int* srcs[2] = {src0, src1};
  const int* dsts[2] = {dst0, dst1};
  float* zs[2]  = {z0, z1};
  float* nss[2] = {ns0, ns1};
  float* nds[2] = {nd0, nd1};
  for (int p = 0; p < 2; ++p) {
    const float* xin = h1;
    for (int s = 0; s < 3; ++s) {
      zero4_kernel<<<cdivl(ND / 4, 256), blk, 0, stream>>>((float4*)agg, ND / 4);
      scatter_kernel<<<cdivl((long)E * 32, 256), blk, 0, stream>>>(
          xin, nss[p], srcs[p], dsts[p], agg, E);
      combine_kernel<<<cdivl((long)N * 32, 256), blk, 0, stream>>>(
          (const float4*)agg, nds[p], (const float4*)h1, (float4*)zs[p], N);
      xin = zs[p];
    }
  }

  // semantic attention  [WMMA fp32, fused tanh + q-dot + reduction]
  attn_kernel<<<cdivl(N, 16), blk, 0, stream>>>(z0, W_att, b_att, q_att, wsum + 0, N);
  attn_kernel<<<cdivl(N, 16), blk, 0, stream>>>(z1, W_att, b_att, q_att, wsum + 1, N);
  softmax2_kernel<<<1, 1, 0, stream>>>(wsum, beta, 1.0f / (float)N);

  // out = beta0*z0 + beta1*z1
  mix_kernel<<<cdivl(ND / 4, 256), blk, 0, stream>>>(
      (const float4*)z0, (const float4*)z1, beta, (float4*)out, ND / 4);
}